// SparseGATConv_65867618452065
// MI455X (gfx1250) — compile-verified
//
#include <hip/hip_runtime.h>
#include <hip/hip_bf16.h>
#include <math.h>

#define N_NODES 50000
#define N_EDGES 800000
#define IN_CH   256
#define OUT_CH  64
#define EPS_F   1e-16f

typedef __attribute__((ext_vector_type(2))) float v2f;
typedef __attribute__((ext_vector_type(8))) float v8f;

// ---------------------------------------------------------------------------
// K0: init accumulators. seg_max=-inf, seg_sum=0, out=0 (out is atomically
// accumulated into, and ws/out are poisoned by the harness).
// ---------------------------------------------------------------------------
__global__ void gat_init(float* __restrict__ seg_max, float* __restrict__ seg_sum,
                         float* __restrict__ out) {
    int i = blockIdx.x * blockDim.x + threadIdx.x;
    if (i < N_NODES) { seg_max[i] = -INFINITY; seg_sum[i] = 0.0f; }
    if (i < N_NODES * OUT_CH) out[i] = 0.0f;
}

// ---------------------------------------------------------------------------
// K1: Wh = x @ W via V_WMMA_F32_16X16X4_F32 (wave32, 16x16 tile per wave).
// A 16x4 f32 layout: lanes 0-15 = M rows, VGPR{0,1} = K{0,1}; lanes 16-31 = K{2,3}.
// B 4x16 f32 layout mirrored; C/D: 8 VGPRs, lane<16 -> M=r, lane>=16 -> M=r+8.
// ---------------------------------------------------------------------------
__global__ __launch_bounds__(128) void gat_gemm_wmma(const float* __restrict__ x,
                                                     const float* __restrict__ W,
                                                     float* __restrict__ Wh) {
    const int wave = threadIdx.x >> 5;     // 0..3 -> column tile of 16
    const int lane = threadIdx.x & 31;
    const int m    = lane & 15;            // M (for A) / N (for B) within tile
    const int hi   = lane >> 4;            // 0: K 0,1   1: K 2,3
    const int row0 = blockIdx.x * 16;      // 50000 = 16*3125, exact
    const int c0   = wave * 16;

    const float* xrow = x + (size_t)(row0 + m) * IN_CH + (hi * 2);
    const float* wcol = W + c0 + m;

    v8f acc = {};
    #pragma unroll 4
    for (int k = 0; k < IN_CH; k += 4) {
        v2f a, b;
        a.x = xrow[k];
        a.y = xrow[k + 1];
        b.x = wcol[(size_t)(k + hi * 2)     * OUT_CH];
        b.y = wcol[(size_t)(k + hi * 2 + 1) * OUT_CH];
        // 8 args: (neg_a, A, neg_b, B, c_mod, C, reuse_a, reuse_b)
        acc = __builtin_amdgcn_wmma_f32_16x16x4_f32(false, a, false, b,
                                                    (short)0, acc, false, false);
    }

    float* o = Wh + (size_t)(row0 + hi * 8) * OUT_CH + c0 + m;
    #pragma unroll
    for (int r = 0; r < 8; ++r) o[(size_t)r * OUT_CH] = acc[r];
}

// ---------------------------------------------------------------------------
// K2: per-node score e[n] = dot(Wh[n,:], a_src + a_dst)
// ---------------------------------------------------------------------------
__global__ void gat_escore(const float* __restrict__ Wh,
                           const float* __restrict__ a_src,
                           const float* __restrict__ a_dst,
                           float* __restrict__ escore) {
    int n = blockIdx.x * blockDim.x + threadIdx.x;
    if (n >= N_NODES) return;
    const float* wr = Wh + (size_t)n * OUT_CH;
    float s = 0.0f;
    #pragma unroll 8
    for (int c = 0; c < OUT_CH; ++c) s += wr[c] * (a_src[c] + a_dst[c]);
    escore[n] = s;
}

// float atomic max via signed/unsigned int ordering trick (init = -inf)
__device__ __forceinline__ void atomicMaxF32(float* addr, float v) {
    if (v >= 0.0f) atomicMax((int*)addr, __float_as_int(v));
    else           atomicMin((unsigned int*)addr, __float_as_uint(v));
}

// ---------------------------------------------------------------------------
// K3: segment max over destination nodes
// ---------------------------------------------------------------------------
__global__ void gat_edge_max(const long long* __restrict__ ei,
                             const float* __restrict__ escore,
                             float* __restrict__ seg_max) {
    int e = blockIdx.x * blockDim.x + threadIdx.x;
    if (e >= N_EDGES) return;
    int r = (int)ei[e];
    int c = (int)ei[N_EDGES + e];
    atomicMaxF32(&seg_max[c], escore[r] + escore[c]);
}

// ---------------------------------------------------------------------------
// K4: ex = exp(e - segmax[col]); segment sum
// ---------------------------------------------------------------------------
__global__ void gat_edge_exp(const long long* __restrict__ ei,
                             const float* __restrict__ escore,
                             const float* __restrict__ seg_max,
                             float* __restrict__ exbuf,
                             float* __restrict__ seg_sum) {
    int e = blockIdx.x * blockDim.x + threadIdx.x;
    if (e >= N_EDGES) return;
    int r = (int)ei[e];
    int c = (int)ei[N_EDGES + e];
    float ex = expf((escore[r] + escore[c]) - seg_max[c]);
    exbuf[e] = ex;
    atomicAdd(&seg_sum[c], ex);
}

// ---------------------------------------------------------------------------
// K5: weighted scatter-add. One wave32 per edge, 2 channels per lane so both
// the Wh-row gather and the output atomics are coalesced across a 64-f32 row.
// Accumulators live in L2 (12.8 MB out << 192 MB).
// ---------------------------------------------------------------------------
__global__ __launch_bounds__(256) void gat_scatter(const long long* __restrict__ ei,
                                                   const float* __restrict__ Wh,
                                                   const float* __restrict__ exbuf,
                                                   const float* __restrict__ seg_sum,
                                                   float* __restrict__ out) {
    int gid  = blockIdx.x * blockDim.x + threadIdx.x;
    int e    = gid >> 5;
    int lane = gid & 31;
    if (e >= N_EDGES) return;
    int r = (int)ei[e];
    int c = (int)ei[N_EDGES + e];
    float att = exbuf[e] / (seg_sum[c] + EPS_F);
    const float* wr = Wh + (size_t)r * OUT_CH;
    float* o = out + (size_t)c * OUT_CH;
    atomicAdd(&o[lane],      att * wr[lane]);
    atomicAdd(&o[lane + 32], att * wr[lane + 32]);
}

// ---------------------------------------------------------------------------
// K6: ELU in place
// ---------------------------------------------------------------------------
__global__ void gat_elu(float* __restrict__ out) {
    int i = blockIdx.x * blockDim.x + threadIdx.x;
    if (i >= N_NODES * OUT_CH) return;
    float v = out[i];
    out[i] = v > 0.0f ? v : (expf(v) - 1.0f);
}

// ---------------------------------------------------------------------------
extern "C" void kernel_launch(void* const* d_in, const int* in_sizes, int n_in,
                              void* d_out, int out_size, void* d_ws, size_t ws_size,
                              hipStream_t stream) {
    const float*     x     = (const float*)d_in[0];
    const long long* ei    = (const long long*)d_in[1];  // int64 edge_index [2, E]
    const float*     W     = (const float*)d_in[2];
    const float*     a_src = (const float*)d_in[3];
    const float*     a_dst = (const float*)d_in[4];
    float* out = (float*)d_out;

    float* ws      = (float*)d_ws;
    float* Wh      = ws;                                   // N*64
    float* escore  = Wh + (size_t)N_NODES * OUT_CH;        // N
    float* seg_max = escore + N_NODES;                     // N
    float* seg_sum = seg_max + N_NODES;                    // N
    float* exbuf   = seg_sum + N_NODES;                    // E

    gat_init   <<<(N_NODES * OUT_CH + 255) / 256, 256, 0, stream>>>(seg_max, seg_sum, out);
    gat_gemm_wmma<<<N_NODES / 16, 128, 0, stream>>>(x, W, Wh);
    gat_escore <<<(N_NODES + 255) / 256, 256, 0, stream>>>(Wh, a_src, a_dst, escore);
    gat_edge_max<<<(N_EDGES + 255) / 256, 256, 0, stream>>>(ei, escore, seg_max);
    gat_edge_exp<<<(N_EDGES + 255) / 256, 256, 0, stream>>>(ei, escore, seg_max, exbuf, seg_sum);
    gat_scatter<<<((size_t)N_EDGES * 32 + 255) / 256, 256, 0, stream>>>(ei, Wh, exbuf, seg_sum, out);
    gat_elu    <<<(N_NODES * OUT_CH + 255) / 256, 256, 0, stream>>>(out);
}